// RoIPoolingLayer_8753143349289
// MI455X (gfx1250) — compile-verified
//
#include <hip/hip_runtime.h>

// RoI bilinear pooling (RoI-Align style), MI455X / gfx1250.
// image: (1,50,50,512) f32, rois: (2000,4) f32 -> out: (1,2000,7,7,512) f32.
// Store-bandwidth bound: ~201 MB out + ~5 MB L2-resident image => ~8.8us floor
// @ 23.3 TB/s. Strategy: 1 block per (roi,py) row, 128 threads (4x wave32).
// For each of the 7 px cells: per-lane async 128b gathers of the 4 source
// pixels into LDS (ASYNCcnt path), double-buffered and software-pipelined via
// partial s_wait_asynccnt (async loads complete in order), blend, and stream
// the result out with non-temporal 128b stores (th:TH_STORE_NT) so the
// write-once output doesn't thrash L2.

#define POOL    7
#define HDIM    50
#define WDIM    50
#define CDIM    512
#define THREADS 128         // CDIM / 4 channels per thread (float4)

typedef float v4f __attribute__((ext_vector_type(4)));

#if defined(__AMDGCN__)
  #define AS1 __attribute__((address_space(1)))
  #define AS3 __attribute__((address_space(3)))
  typedef int v4i __attribute__((vector_size(16)));
  #if __has_builtin(__builtin_amdgcn_global_load_async_to_lds_b128)
    #define ASYNC_MODE 1    // clang builtin path (confirmed present in round 1)
  #else
    #define ASYNC_MODE 2    // inline-asm path (CDNA5 ISA)
  #endif
#else
  #define ASYNC_MODE 0      // host pass: never executed
#endif

__device__ __forceinline__ void async_gather16(const float* g, float* l)
{
#if ASYNC_MODE == 1
    __builtin_amdgcn_global_load_async_to_lds_b128((AS1 v4i*)g, (AS3 v4i*)l,
                                                   /*offset=*/0, /*cpol=*/0);
#elif ASYNC_MODE == 2
    unsigned lds_off = (unsigned)(unsigned long long)(AS3 void*)l;
    asm volatile("global_load_async_to_lds_b128 %0, %1, off"
                 :: "v"(lds_off), "v"(g) : "memory");
#else
    (void)g; (void)l;
#endif
}

template <int N>
__device__ __forceinline__ void wait_async()
{
#if ASYNC_MODE == 1 && __has_builtin(__builtin_amdgcn_s_wait_asynccnt)
    __builtin_amdgcn_s_wait_asynccnt(N);
#elif ASYNC_MODE
    asm volatile("s_wait_asynccnt %0" :: "i"(N) : "memory");
#endif
}

// Mirrors reference _axis_coords for one sample point i.
// jnp.round -> rintf (RNE); .astype(int32) -> trunc toward zero (values >= 0);
// jnp.clip -> min/max. `scale` = size/POOL precomputed once per axis.
__device__ __forceinline__ void axis_coords(float start, float size, float scale,
                                            int i, int limit,
                                            int& i0, int& i1, float& frac)
{
    float loc = ((float)i + 0.5f) * scale - 0.5f;
    loc = fminf(fmaxf(loc, 0.0f), size - 1.0f);
    float lo = floorf(loc);
    frac = loc - lo;
    int s = (int)start;
    i0 = (int)lo + s;
    i1 = (int)fminf(lo + 1.0f, size - 1.0f) + s;
    i0 = min(max(i0, 0), limit - 1);
    i1 = min(max(i1, 0), limit - 1);
}

__global__ __launch_bounds__(THREADS) void roi_align_kernel(
    const float* __restrict__ img,
    const float* __restrict__ rois,
    float* __restrict__ out)
{
    // Double-buffered staging: 2 x (4 pixels x 512 ch) = 16 KB.
    __shared__ __align__(16) float tile[2][4 * CDIM];

    const int bid = blockIdx.x;          // bid = n*7 + py
    const int n   = bid / POOL;
    const int py  = bid - n * POOL;
    const int ch  = (int)threadIdx.x * 4;

    // rois row is uniform across the block -> scalar loads / scalar float ops.
    const float4 q = *reinterpret_cast<const float4*>(rois + (size_t)n * 4);
    const float r = rintf(q.x * 0.0625f);   // q = round(roi / 16)
    const float c = rintf(q.y * 0.0625f);
    const float w = rintf(q.z * 0.0625f);
    const float h = rintf(q.w * 0.0625f);
    const float sw = w / (float)POOL;       // one divide per axis per block
    const float sh = h / (float)POOL;

    int y0, y1; float fy;
    axis_coords(c, h, sh, py, HDIM, y0, y1, fy);          // y-axis: start=c, size=h

    int x0a[POOL], x1a[POOL]; float fxa[POOL];
#pragma unroll
    for (int px = 0; px < POOL; ++px)                     // x-axis: start=r, size=w
        axis_coords(r, w, sw, px, WDIM, x0a[px], x1a[px], fxa[px]);

    // Per-thread bases for the two source rows (channel slice ch..ch+3).
    const float* row0 = img + (size_t)y0 * (WDIM * CDIM) + ch;
    const float* row1 = img + (size_t)y1 * (WDIM * CDIM) + ch;
    const float wy0 = 1.0f - fy;

    // Each lane reads back exactly the bytes it gathered -> no barrier, only
    // ASYNCcnt waits. Async loads complete in order, so with 8 outstanding,
    // s_wait_asynccnt 4 guarantees the oldest 4 (current cell) have landed.
#if ASYNC_MODE
    // Prologue: issue gathers for px = 0 into buffer 0.
    async_gather16(row0 + (size_t)x0a[0] * CDIM, &tile[0][0 * CDIM + ch]);
    async_gather16(row0 + (size_t)x1a[0] * CDIM, &tile[0][1 * CDIM + ch]);
    async_gather16(row1 + (size_t)x0a[0] * CDIM, &tile[0][2 * CDIM + ch]);
    async_gather16(row1 + (size_t)x1a[0] * CDIM, &tile[0][3 * CDIM + ch]);
#endif

#pragma unroll
    for (int px = 0; px < POOL; ++px) {
        const int b = px & 1;
        v4f v00, v01, v10, v11;
#if ASYNC_MODE
        if (px + 1 < POOL) {
            const int nb = (px + 1) & 1;
            async_gather16(row0 + (size_t)x0a[px + 1] * CDIM, &tile[nb][0 * CDIM + ch]);
            async_gather16(row0 + (size_t)x1a[px + 1] * CDIM, &tile[nb][1 * CDIM + ch]);
            async_gather16(row1 + (size_t)x0a[px + 1] * CDIM, &tile[nb][2 * CDIM + ch]);
            async_gather16(row1 + (size_t)x1a[px + 1] * CDIM, &tile[nb][3 * CDIM + ch]);
            wait_async<4>();   // drain current cell's 4, keep next cell's in flight
        } else {
            wait_async<0>();
        }
        v00 = *reinterpret_cast<const v4f*>(&tile[b][0 * CDIM + ch]);
        v01 = *reinterpret_cast<const v4f*>(&tile[b][1 * CDIM + ch]);
        v10 = *reinterpret_cast<const v4f*>(&tile[b][2 * CDIM + ch]);
        v11 = *reinterpret_cast<const v4f*>(&tile[b][3 * CDIM + ch]);
#else
        (void)tile; (void)b;
        v00 = *reinterpret_cast<const v4f*>(row0 + (size_t)x0a[px] * CDIM);
        v01 = *reinterpret_cast<const v4f*>(row0 + (size_t)x1a[px] * CDIM);
        v10 = *reinterpret_cast<const v4f*>(row1 + (size_t)x0a[px] * CDIM);
        v11 = *reinterpret_cast<const v4f*>(row1 + (size_t)x1a[px] * CDIM);
#endif
        // Reference form: top = a*(1-fx)+b*fx; out = top*(1-fy)+bot*fy.
        const float fx  = fxa[px];
        const float wx0 = 1.0f - fx;
        v4f res = (v00 * wx0 + v01 * fx) * wy0 + (v10 * wx0 + v11 * fx) * fy;

        // out[((n*7+py)*7+px)*512 + ch] == out[(bid*7+px)*512 + ch].
        // Write-once stream -> non-temporal (th:TH_STORE_NT), keep L2 for image.
        float* dst = out + ((size_t)bid * POOL + px) * CDIM + ch;
        __builtin_nontemporal_store(res, reinterpret_cast<v4f*>(dst));
    }
}

extern "C" void kernel_launch(void* const* d_in, const int* in_sizes, int n_in,
                              void* d_out, int out_size, void* d_ws, size_t ws_size,
                              hipStream_t stream)
{
    (void)n_in; (void)out_size; (void)d_ws; (void)ws_size;
    const float* img  = (const float*)d_in[0];
    const float* rois = (const float*)d_in[1];
    float* out = (float*)d_out;
    const int n_rois = in_sizes[1] / 4;              // rois: (N,4)
    dim3 grid((unsigned)(n_rois * POOL));            // one block per (roi, py)
    roi_align_kernel<<<grid, THREADS, 0, stream>>>(img, rois, out);
}